// CrossBlockAttention_51384988729525
// MI455X (gfx1250) — compile-verified
//
#include <hip/hip_runtime.h>
#include <math.h>

#define D_MODEL  256
#define N_HEADS  8
#define HEAD_DIM 32
#define SEQ      2048
#define BATCH    4
#define ROWS     (BATCH * SEQ)           // 8192
#define SCALEF   0.17677669529663687f    // HEAD_DIM^-0.5
#define LSTRIDE  (SEQ + 4)               // LDS row stride (floats), 16B-aligned, bank-skewed

typedef __attribute__((ext_vector_type(2))) float v2f;
typedef __attribute__((ext_vector_type(4))) float v4f;
typedef __attribute__((ext_vector_type(8))) float v8f;

static __device__ __forceinline__ v8f wmma_f32(v2f a, v2f b, v8f c) {
  // D = A(16x4,f32) * B(4x16,f32) + C(16x16,f32)
  return __builtin_amdgcn_wmma_f32_16x16x4_f32(false, a, false, b, (short)0, c,
                                               false, false);
}

// ---------------------------------------------------------------------------
// Kernel 1: Q/K/V projections.  y = x @ W.T + b, stored head-major [B,H,N,32].
// One wave computes one 16x16 output tile; grid.z selects Wq/Wk/Wv.
// ---------------------------------------------------------------------------
__global__ __launch_bounds__(128) void qkv_kernel(
    const float* __restrict__ x,
    const float* __restrict__ Wq, const float* __restrict__ bq,
    const float* __restrict__ Wk, const float* __restrict__ bk,
    const float* __restrict__ Wv, const float* __restrict__ bv,
    float* __restrict__ Qo, float* __restrict__ Ko, float* __restrict__ Vo) {
  const int lane = threadIdx.x & 31;
  const int wave = threadIdx.x >> 5;
  const int lr = lane & 15, hi = lane >> 4;
  const int m0 = blockIdx.x * 16;
  const int n0 = (blockIdx.y * 4 + wave) * 16;
  const int z = blockIdx.z;
  const float* W    = (z == 0) ? Wq : (z == 1) ? Wk : Wv;
  const float* bias = (z == 0) ? bq : (z == 1) ? bk : bv;
  float*       Y    = (z == 0) ? Qo : (z == 1) ? Ko : Vo;

  // A lane layout (16x4 f32): lane<16 -> K=k,k+1 ; lane>=16 -> K=k+2,k+3
  const float* ap = x + (size_t)(m0 + lr) * D_MODEL + 2 * hi;
  // B[k][n] = W[n][k] (W.T): same pairing on the K axis, N = lane&15
  const float* bp = W + (size_t)(n0 + lr) * D_MODEL + 2 * hi;
  v8f c = {};
#pragma unroll 8
  for (int k = 0; k < D_MODEL; k += 4) {
    v2f a = *(const v2f*)(ap + k);
    v2f b = *(const v2f*)(bp + k);
    c = wmma_f32(a, b, c);
  }
  const float bb = bias[n0 + lr];
  const int o = n0 + lr;
  const int h = o >> 5, d = o & (HEAD_DIM - 1);
#pragma unroll
  for (int r = 0; r < 8; r++) {
    const int m = m0 + r + 8 * hi;            // C layout: VGPR r -> M=r / M=r+8
    const int b_ = m >> 11, n = m & (SEQ - 1);
    Y[(((size_t)b_ * N_HEADS + h) * SEQ + n) * HEAD_DIM + d] = c[r] + bb;
  }
}

// ---------------------------------------------------------------------------
// Kernel 2: per (b,h, 16-query strip): WMMA scores into 131KB LDS tile,
// iterative top-16 per row, softmax, zero+scatter weight rows straight to
// global (non-temporal), sparse attn @ V into workspace.
// ---------------------------------------------------------------------------
__global__ __launch_bounds__(128) void attn_kernel(
    const float* __restrict__ Q, const float* __restrict__ K,
    const float* __restrict__ V, const unsigned char* __restrict__ mask,
    float* __restrict__ attnw, float* __restrict__ aout) {
  const int bh = blockIdx.x;        // b*8 + h
  const int q0 = blockIdx.y * 16;
  const int b = bh >> 3, h = bh & 7;
  const int lane = threadIdx.x & 31;
  const int wave = threadIdx.x >> 5;
  const int lr = lane & 15, hi = lane >> 4;

  extern __shared__ float lds[];
  float* sc   = lds;                    // [16][LSTRIDE] score tile
  float* topv = lds + 16 * LSTRIDE;     // [16][16] top-k values
  float* topw = topv + 256;             // [16][16] softmax weights
  int*   topi = (int*)(topw + 256);     // [16][16] top-k key indices

  const float* Qb = Q + ((size_t)bh * SEQ + q0) * HEAD_DIM;
  const float* Kb = K + (size_t)bh * SEQ * HEAD_DIM;
  const float* Vb = V + (size_t)bh * SEQ * HEAD_DIM;

  // Pre-scaled Q fragments held in registers, reused across all key tiles.
  v2f afrag[8];
#pragma unroll
  for (int kk = 0; kk < 8; kk++) {
    const float* p = Qb + lr * HEAD_DIM + kk * 4 + 2 * hi;
    v2f a; a.x = p[0] * SCALEF; a.y = p[1] * SCALEF;
    afrag[kk] = a;
  }

  // Scores: each wave covers 512 keys (32 tiles of 16), 8 WMMA per tile.
  for (int i = 0; i < 32; i++) {
    const int n0 = (i * 4 + wave) * 16;
    const float* bp = Kb + (size_t)(n0 + lr) * HEAD_DIM + 2 * hi;
    if (i < 31)  // prefetch next K tile (+64 keys = +2048 floats) into caches
      __builtin_prefetch(bp + 64 * HEAD_DIM, 0, 3);
    v8f c = {};
#pragma unroll
    for (int kk = 0; kk < 8; kk++) {
      v2f bfrag = *(const v2f*)(bp + kk * 4);
      c = wmma_f32(afrag[kk], bfrag, c);
    }
#pragma unroll
    for (int r = 0; r < 8; r++)
      sc[(r + 8 * hi) * LSTRIDE + n0 + lr] = c[r];
  }
  __syncthreads();

  // Top-16 per query row (wave handles 4 rows), then softmax with key mask
  // applied AFTER selection (matching reference order).
  const unsigned char* mrow = mask + (size_t)b * SEQ;
  for (int j = 0; j < 4; j++) {
    const int q = wave * 4 + j;
    float* row = sc + q * LSTRIDE;
    for (int t = 0; t < 16; t++) {
      float bv = -INFINITY; int bi = 0x7fffffff;
      for (int kidx = lane; kidx < SEQ; kidx += 32) {
        float v = row[kidx];
        if (v > bv || (v == bv && kidx < bi)) { bv = v; bi = kidx; }
      }
#pragma unroll
      for (int off = 16; off > 0; off >>= 1) {
        float ov = __shfl_xor(bv, off, 32);
        int   oi = __shfl_xor(bi, off, 32);
        if (ov > bv || (ov == bv && oi < bi)) { bv = ov; bi = oi; }
      }
      if (lane == 0) { topv[q * 16 + t] = bv; topi[q * 16 + t] = bi; }
      row[bi] = -INFINITY;   // bi is wave-uniform; knock out for next round
    }
    float vl = -INFINITY; bool valid = false;
    if (lane < 16) {
      vl = topv[q * 16 + lane];
      valid = (mrow[topi[q * 16 + lane]] != 0);
    }
    float mv = valid ? vl : -INFINITY;
    float mx = mv;
#pragma unroll
    for (int off = 16; off > 0; off >>= 1)
      mx = fmaxf(mx, __shfl_xor(mx, off, 32));
    float e = valid ? __expf(mv - mx) : 0.0f;
    float s = e;
#pragma unroll
    for (int off = 16; off > 0; off >>= 1)
      s += __shfl_xor(s, off, 32);
    float w = (s > 0.0f) ? (e / s) : 0.0f;
    if (lane < 16) topw[q * 16 + lane] = w;
  }
  __syncthreads();

  // Weight rows = zeros + 16 scattered softmax weights. Write zeros straight
  // to global with NT hint (write-once stream, 536MB total: keep it out of
  // the 192MB L2 so x/Q/K/V/W stay resident), then scatter after the barrier
  // (which drains store counters, ordering zero-fill before the scatter).
  float* gout = attnw + ((size_t)bh * SEQ + q0) * SEQ;   // contiguous [16][2048]
  const v4f z4 = {0.f, 0.f, 0.f, 0.f};
  for (int e = threadIdx.x; e < 16 * SEQ / 4; e += 128)
    __builtin_nontemporal_store(z4, (v4f*)gout + e);
  __syncthreads();
  for (int e = threadIdx.x; e < 256; e += 128) {
    const int q = e >> 4;
    __builtin_nontemporal_store(topw[e], gout + (size_t)q * SEQ + topi[e]);
  }

  // Sparse attn @ V: 16 weighted V rows per query; lane == head dim.
  for (int j = 0; j < 4; j++) {
    const int q = wave * 4 + j;
    float acc = 0.f;
#pragma unroll
    for (int t = 0; t < 16; t++) {
      const float w = topw[q * 16 + t];
      const int  ki = topi[q * 16 + t];
      acc += w * Vb[(size_t)ki * HEAD_DIM + lane];
    }
    aout[(size_t)(b * SEQ + q0 + q) * D_MODEL + h * HEAD_DIM + lane] = acc;
  }
}

// ---------------------------------------------------------------------------
// Kernel 3: output projection out = attn @ Wo.T + bo, row-major [B*N, 256].
// ---------------------------------------------------------------------------
__global__ __launch_bounds__(128) void outproj_kernel(
    const float* __restrict__ A, const float* __restrict__ Wo,
    const float* __restrict__ bo, float* __restrict__ out) {
  const int lane = threadIdx.x & 31;
  const int wave = threadIdx.x >> 5;
  const int lr = lane & 15, hi = lane >> 4;
  const int m0 = blockIdx.x * 16;
  const int n0 = (blockIdx.y * 4 + wave) * 16;
  const float* ap = A  + (size_t)(m0 + lr) * D_MODEL + 2 * hi;
  const float* bp = Wo + (size_t)(n0 + lr) * D_MODEL + 2 * hi;
  v8f c = {};
#pragma unroll 8
  for (int k = 0; k < D_MODEL; k += 4) {
    v2f a = *(const v2f*)(ap + k);
    v2f b = *(const v2f*)(bp + k);
    c = wmma_f32(a, b, c);
  }
  const float bb = bo[n0 + lr];
#pragma unroll
  for (int r = 0; r < 8; r++)
    out[(size_t)(m0 + r + 8 * hi) * D_MODEL + n0 + lr] = c[r] + bb;
}

// ---------------------------------------------------------------------------
extern "C" void kernel_launch(void* const* d_in, const int* in_sizes, int n_in,
                              void* d_out, int out_size, void* d_ws, size_t ws_size,
                              hipStream_t stream) {
  const float* x  = (const float*)d_in[0];
  const unsigned char* mask = (const unsigned char*)d_in[1];  // jnp bool -> 1B
  const float* Wq = (const float*)d_in[2];
  const float* bq = (const float*)d_in[3];
  const float* Wk = (const float*)d_in[4];
  const float* bk = (const float*)d_in[5];
  const float* Wv = (const float*)d_in[6];
  const float* bv = (const float*)d_in[7];
  const float* Wo = (const float*)d_in[8];
  const float* bo = (const float*)d_in[9];

  const size_t QKVSZ = (size_t)ROWS * D_MODEL;  // 2,097,152 floats each
  float* Qws = (float*)d_ws;
  float* Kws = Qws + QKVSZ;
  float* Vws = Kws + QKVSZ;
  float* Aws = Vws + QKVSZ;                      // attn output [B*N, 256]

  float* out   = (float*)d_out;                  // [B,N,256]
  float* attnw = out + QKVSZ;                    // [B,H,N,N] follows in d_out

  qkv_kernel<<<dim3(ROWS / 16, 4, 3), 128, 0, stream>>>(
      x, Wq, bq, Wk, bk, Wv, bv, Qws, Kws, Vws);

  const size_t ldsBytes = (size_t)16 * LSTRIDE * sizeof(float) +
                          3 * 256 * sizeof(float);   // ~134 KB of 320 KB WGP LDS
  attn_kernel<<<dim3(BATCH * N_HEADS, SEQ / 16), 128, ldsBytes, stream>>>(
      Qws, Kws, Vws, mask, attnw, Aws);

  outproj_kernel<<<dim3(ROWS / 16, 4), 128, 0, stream>>>(Aws, Wo, bo, out);
}